// VectorQuantizer_33380485824874
// MI455X (gfx1250) — compile-verified
//
#include <hip/hip_runtime.h>

typedef __attribute__((ext_vector_type(2))) float v2f;
typedef __attribute__((ext_vector_type(8))) float v8f;
typedef __attribute__((ext_vector_type(4))) int   v4i;

#define VQ_D            256
#define VQ_K            1024
#define VQ_NROWS        65536          // 16*4096 rows of length 256
#define ROWS_PER_WAVE   16
#define WAVES_PER_BLOCK 4
#define ROWS_PER_BLOCK  (ROWS_PER_WAVE * WAVES_PER_BLOCK)   // 64
#define KTILE           16
#define NTILES          (VQ_K / KTILE)                       // 64
#define CS_PAD          4
#define PIPE            8                                    // B double-buffer depth
#define NGROUPS         (VQ_D / 4 / PIPE)                    // 8 groups of 8 WMMAs
#define TOTAL_WAVES     ((VQ_NROWS / ROWS_PER_BLOCK) * WAVES_PER_BLOCK) // 4096
#define LOSS_SCALE      (1.25f / (float)((size_t)VQ_NROWS * VQ_D))

// Async global->LDS staging path (gfx1250 GLOBAL_LOAD_ASYNC_TO_LDS_B128),
// guarded so host pass / older toolchains fall back to sync staging.
#if defined(__gfx1250__) && __has_builtin(__builtin_amdgcn_global_load_async_to_lds_b128) && \
    __has_builtin(__builtin_amdgcn_s_wait_asynccnt)
#define VQ_ASYNC_STAGE 1
typedef __attribute__((address_space(1))) v4i g_v4i;   // global (AS1) int32x4
typedef __attribute__((address_space(3))) v4i l_v4i;   // LDS (AS3) int32x4
#else
#define VQ_ASYNC_STAGE 0
#endif

// ---------------------------------------------------------------------------
// c_sq[n] = ||codeword_n||^2
// ---------------------------------------------------------------------------
__global__ void vq_csq_kernel(const float* __restrict__ cw, float* __restrict__ csq) {
    int n = blockIdx.x * blockDim.x + threadIdx.x;
    if (n >= VQ_K) return;
    const float4* p = (const float4*)(cw + (size_t)n * VQ_D);
    float s = 0.f;
#pragma unroll
    for (int i = 0; i < VQ_D / 4; ++i) {
        float4 v = p[i];
        s += v.x * v.x + v.y * v.y + v.z * v.z + v.w * v.w;
    }
    csq[n] = s;
}

// ---------------------------------------------------------------------------
// Main VQ kernel: per wave, 16 rows of X vs all 1024 codewords via
// V_WMMA_F32_16X16X4_F32 (exact f32), async-staged codeword tiles in LDS,
// double-buffered B operands, running argmin on (c_sq - 2*dot), gather of
// winners, quantized + index outputs, per-wave loss partial.
// ---------------------------------------------------------------------------
__global__ __launch_bounds__(WAVES_PER_BLOCK * 32, 1)
void vq_main_kernel(const float* __restrict__ X, const float* __restrict__ CW,
                    const float* __restrict__ CSQ,
                    float* __restrict__ outQ, float* __restrict__ outIdx,
                    float* __restrict__ lossPartials) {
    __shared__ float Cs[KTILE][VQ_D + CS_PAD];

    const int tid  = threadIdx.x;
    const int lane = tid & 31;
    const int wave = tid >> 5;
    const int half = lane >> 4;          // selects K-pair in A/B layouts
    const int l15  = lane & 15;
    const int rowbase = blockIdx.x * ROWS_PER_BLOCK + wave * ROWS_PER_WAVE;

    // --- preload A-tile (16 x 256) into registers, WMMA 16x16x4 f32 A layout:
    // lane L holds row m = L&15; the two A VGPRs hold K = 2*half, 2*half+1.
    v2f a[VQ_D / 4];
    {
        const float* xrow = X + (size_t)(rowbase + l15) * VQ_D + 2 * half;
#pragma unroll
        for (int j = 0; j < VQ_D / 4; ++j)
            a[j] = *(const v2f*)(xrow + 4 * j);
    }

    float minval[8];
    int   minidx[8];
#pragma unroll
    for (int v = 0; v < 8; ++v) { minval[v] = 3.4e38f; minidx[v] = 0; }

    for (int ct = 0; ct < NTILES; ++ct) {
        // ---- stage 16-codeword tile into LDS ----
        __syncthreads();
        {
            const int r = tid >> 3;              // 0..15 (8 threads per row)
            const int c = (tid & 7) * 4;
            const float* src = CW + (size_t)(ct * KTILE + r) * VQ_D;
#if VQ_ASYNC_STAGE
            // ASYNCcnt-tracked DMA into LDS, no VGPR round-trip.
#pragma unroll
            for (int v = 0; v < 8; ++v)
                __builtin_amdgcn_global_load_async_to_lds_b128(
                    (g_v4i*)(src + c + v * 32),
                    (l_v4i*)&Cs[r][c + v * 32], 0, 0);
            __builtin_amdgcn_s_wait_asynccnt(0);  // LDS writes done before barrier
#else
#pragma unroll
            for (int v = 0; v < 8; ++v)
                *(float4*)&Cs[r][c + v * 32] = *(const float4*)(src + c + v * 32);
#endif
        }
        if (tid == 0 && ct + 1 < NTILES)  // CDNA5 global_prefetch_b8 on next tile
            __builtin_prefetch(CW + (size_t)(ct + 1) * KTILE * VQ_D, 0, 1);
        __syncthreads();

        const float csq = CSQ[ct * KTILE + l15];

        // ---- 64x V_WMMA_F32_16X16X4_F32 with software-pipelined B ----
        // B element for step j lives at Cs[l15][4*j + 2*half]; conflict-free
        // banks (4n + k) mod 64 across the 16 lanes of each half.
        const float* bsrc = &Cs[l15][2 * half];

        v2f breg[2][PIPE];
#pragma unroll
        for (int u = 0; u < PIPE; ++u)
            breg[0][u] = *(const v2f*)(bsrc + 4 * u);

        v8f acc = {};
#pragma unroll
        for (int g = 0; g < NGROUPS; ++g) {
            const int cur = g & 1;
            if (g + 1 < NGROUPS) {               // issue next batch's DS loads
#pragma unroll
                for (int u = 0; u < PIPE; ++u)
                    breg[cur ^ 1][u] = *(const v2f*)(bsrc + 4 * ((g + 1) * PIPE + u));
            }
#pragma unroll
            for (int u = 0; u < PIPE; ++u)
                acc = __builtin_amdgcn_wmma_f32_16x16x4_f32(
                    false, a[g * PIPE + u], false, breg[cur][u],
                    (short)0, acc, false, false);
        }

        // D element (m, n): lane holds rows m = v + 8*half, column n = lane&15
        const int n_g = ct * KTILE + l15;
#pragma unroll
        for (int v = 0; v < 8; ++v) {
            float score = csq - 2.0f * acc[v];   // argmin-equivalent distance
            if (score < minval[v]) { minval[v] = score; minidx[v] = n_g; }
        }
    }

    // ---- argmin reduce across the 16 lanes of each half-wave ----
#pragma unroll
    for (int mask = 1; mask < 16; mask <<= 1) {
#pragma unroll
        for (int v = 0; v < 8; ++v) {
            float ov = __shfl_xor(minval[v], mask, 16);
            int   oi = __shfl_xor(minidx[v], mask, 16);
            if (ov < minval[v] || (ov == minval[v] && oi < minidx[v])) {
                minval[v] = ov; minidx[v] = oi;
            }
        }
    }

    // ---- write encoding indices (one lane per row, cast to float) ----
    if (l15 < 8) {
        int m = (lane & 7) + 8 * half;
        outIdx[rowbase + m] = (float)minidx[lane & 7];
    }

    // ---- gather winning codewords, store quantized, accumulate loss ----
    float lossAcc = 0.f;
#pragma unroll
    for (int m = 0; m < 16; ++m) {
        const int idxm = __shfl(minidx[m & 7], (m >> 3) * 16, 32);
        const float4* cw4 = (const float4*)(CW + (size_t)idxm * VQ_D);
        const float4* x4  = (const float4*)(X + (size_t)(rowbase + m) * VQ_D);
        float4*       o4  = (float4*)(outQ + (size_t)(rowbase + m) * VQ_D);
#pragma unroll
        for (int p = 0; p < 2; ++p) {
            const int e = lane * 2 + p;          // 64 float4s per row
            float4 q = cw4[e];
            float4 x = x4[e];
            o4[e] = q;
            float dx = q.x - x.x, dy = q.y - x.y, dz = q.z - x.z, dw = q.w - x.w;
            lossAcc += dx * dx + dy * dy + dz * dz + dw * dw;
        }
    }
#pragma unroll
    for (int mask = 16; mask >= 1; mask >>= 1)
        lossAcc += __shfl_xor(lossAcc, mask, 32);
    if (lane == 0)
        lossPartials[blockIdx.x * WAVES_PER_BLOCK + wave] = lossAcc;
}

// ---------------------------------------------------------------------------
// Deterministic fixed-order loss reduction (no float atomics).
// ---------------------------------------------------------------------------
__global__ void vq_loss_reduce_kernel(const float* __restrict__ partials,
                                      float* __restrict__ outLoss) {
    __shared__ float red[256];
    const int t = threadIdx.x;
    float s = 0.f;
    for (int i = t; i < TOTAL_WAVES; i += 256) s += partials[i];
    red[t] = s;
    __syncthreads();
    for (int off = 128; off > 0; off >>= 1) {
        if (t < off) red[t] += red[t + off];
        __syncthreads();
    }
    if (t == 0) *outLoss = red[0] * LOSS_SCALE;
}

// ---------------------------------------------------------------------------
extern "C" void kernel_launch(void* const* d_in, const int* in_sizes, int n_in,
                              void* d_out, int out_size, void* d_ws, size_t ws_size,
                              hipStream_t stream) {
    (void)in_sizes; (void)n_in; (void)out_size; (void)ws_size;
    const float* X  = (const float*)d_in[0];     // (16,4096,256) f32
    const float* CW = (const float*)d_in[1];     // (1024,256) f32

    float* out     = (float*)d_out;
    float* outQ    = out;                                 // 16777216 floats
    float* outIdx  = out + (size_t)VQ_NROWS * VQ_D;       // 65536 floats
    float* outLoss = outIdx + VQ_NROWS;                   // 1 float

    float* csq      = (float*)d_ws;                       // 1024 floats
    float* partials = csq + VQ_K;                         // 4096 floats

    vq_csq_kernel<<<(VQ_K + 255) / 256, 256, 0, stream>>>(CW, csq);
    vq_main_kernel<<<VQ_NROWS / ROWS_PER_BLOCK, WAVES_PER_BLOCK * 32, 0, stream>>>(
        X, CW, csq, outQ, outIdx, partials);
    vq_loss_reduce_kernel<<<1, 256, 0, stream>>>(partials, outLoss);
}